// AttentionalPropagation_55362128445841
// MI455X (gfx1250) — compile-verified
//
#include <hip/hip_runtime.h>
#include <hip/hip_bf16.h>

typedef __attribute__((ext_vector_type(2))) float v2f;
typedef __attribute__((ext_vector_type(8))) float v8f;

// D = A(16x4) * B(4x16) + C(16x16), fp32, wave32
#define WMMA_F32X4(a, b, c) \
  __builtin_amdgcn_wmma_f32_16x16x4_f32(false, (a), false, (b), (short)0, (c), false, false)

#define GLOBAL_AS __attribute__((address_space(1)))
#define LDS_AS    __attribute__((address_space(3)))

#if __has_builtin(__builtin_amdgcn_global_load_async_to_lds_b32)
#define HAVE_ASYNC_LDS 1
#else
#define HAVE_ASYNC_LDS 0
#endif

__device__ __forceinline__ void async_copy_b32(const float* gsrc, float* lds_dst) {
#if HAVE_ASYNC_LDS
  __builtin_amdgcn_global_load_async_to_lds_b32(
      (GLOBAL_AS int*)gsrc, (LDS_AS int*)lds_dst, 0, 0);
#else
  *lds_dst = *gsrc;
#endif
}

__device__ __forceinline__ void wait_async_lds() {
#if HAVE_ASYNC_LDS
#if __has_builtin(__builtin_amdgcn_s_wait_asynccnt)
  __builtin_amdgcn_s_wait_asynccnt(0);
#else
  asm volatile("s_wait_asynccnt 0" ::: "memory");
#endif
#endif
}

namespace {
constexpr int BATCH = 2;
constexpr int D     = 256;
constexpr int N     = 4096;
constexpr int H     = 4;
constexpr int D2    = 512;   // 2*D
}

// ---------------------------------------------------------------------------
// Generic conv1 (channel-wise linear): Y[b] = W (OxI) * X[b] (IxN) + bias.
// X rows [0, split) come from X0, rows [split, I) from X1 (for the concat).
// Grid: (N/128, O/16, BATCH), block 256 (8 waves, each one 16x16 tile).
// ---------------------------------------------------------------------------
__global__ __launch_bounds__(256)
void conv1_gemm_kernel(const float* __restrict__ W, const float* __restrict__ bias,
                       const float* __restrict__ X0, const float* __restrict__ X1,
                       int split, int O, int I, float* __restrict__ Y)
{
  constexpr int LDW = 514;               // max I (512) + 2 pad, keeps 8B align
  __shared__ float lds_w[16 * LDW];      // 32.9 KB

  const int t    = threadIdx.x;
  const int wave = t >> 5;
  const int lane = t & 31;
  const int half = lane >> 4;
  const int col  = lane & 15;
  const int b    = blockIdx.z;
  const int o0   = blockIdx.y * 16;
  const int n0   = blockIdx.x * 128 + wave * 16;

  // Stage the 16 x I weight tile once (all waves share the A operand).
  for (int e = t; e < 16 * I; e += 256) {
    int r = e / I, i = e - r * I;
    lds_w[r * LDW + i] = W[(size_t)(o0 + r) * I + i];
  }
  __syncthreads();

  // Per-lane column base pointers for the two activation sources.
  const float* p0 = X0 + (size_t)b * split * N + n0 + col;
  const float* p1 = X1 + (size_t)b * (I - split) * N + n0 + col;
  const float* wrow = &lds_w[col * LDW + 2 * half];

  v8f acc = {};
  // k in [0, split): source X0   (split is a multiple of 4, uniform per wave)
  for (int k0 = 0; k0 < split; k0 += 4) {
    v2f a; a.x = wrow[k0]; a.y = wrow[k0 + 1];
    const int i = k0 + 2 * half;
    v2f bf;
    bf.x = p0[(size_t)i * N];
    bf.y = p0[(size_t)(i + 1) * N];
    acc = WMMA_F32X4(a, bf, acc);
  }
  // k in [split, I): source X1
  for (int k0 = split; k0 < I; k0 += 4) {
    v2f a; a.x = wrow[k0]; a.y = wrow[k0 + 1];
    const int i = (k0 - split) + 2 * half;
    v2f bf;
    bf.x = p1[(size_t)i * N];
    bf.y = p1[(size_t)(i + 1) * N];
    acc = WMMA_F32X4(a, bf, acc);
  }

  float* yb = Y + (size_t)b * O * N;
#pragma unroll
  for (int r = 0; r < 8; ++r) {
    const int o = o0 + r + 8 * half;             // C layout: rows r / r+8
    yb[(size_t)o * N + n0 + col] = acc[r] + bias[o];
  }
}

// ---------------------------------------------------------------------------
// Flash attention, fp32 WMMA. Head-dim 64, scale 1/8.
// q/k/v/attn are (B, D, N); head h owns channels d*H + h, d in [0,64).
// Grid: (N/128, H, BATCH), block 256; each wave owns a 16-query tile and
// streams K/V in 64-wide blocks staged in LDS via ASYNCcnt-tracked
// global_load_async_to_lds (the per-lane LDS address does the K transpose).
// ---------------------------------------------------------------------------
__global__ __launch_bounds__(256)
void attention_kernel(const float* __restrict__ q, const float* __restrict__ k,
                      const float* __restrict__ v, float* __restrict__ attn)
{
  constexpr int LKV = 66;                    // 64 + 2 pad (even -> 8B align)
  __shared__ float ldsK[64 * LKV];           // K^T block: [m][d]
  __shared__ float ldsV[64 * LKV];           // V   block: [d][m]
  __shared__ float ldsP[8 * 16 * 17];        // per-wave P tile [m][n]

  const int t    = threadIdx.x;
  const int wave = t >> 5;
  const int lane = t & 31;
  const int half = lane >> 4;
  const int col  = lane & 15;
  const int h    = blockIdx.y;
  const int b    = blockIdx.z;
  const int n0   = blockIdx.x * 128 + wave * 16;

  const float* qb = q + (size_t)b * D * N;
  const float* kb = k + (size_t)b * D * N;
  const float* vb = v + (size_t)b * D * N;

  // Resident Q B-fragments: Qf[kk] covers d = 4*kk .. 4*kk+3 at 16 n-columns.
  v2f Qf[16];
#pragma unroll
  for (int kk = 0; kk < 16; ++kk) {
    const int d = 4 * kk + 2 * half;
    Qf[kk].x = qb[((size_t)d * H + h) * N + n0 + col];
    Qf[kk].y = qb[((size_t)(d + 1) * H + h) * N + n0 + col];
  }

  v8f acc[4] = {};                            // O tile: 64 d-rows x 16 n-cols
  float m_run = -1e30f, l_run = 0.0f;
  float* Pw = &ldsP[wave * 16 * 17];

  for (int mb = 0; mb < N; mb += 64) {
    __syncthreads();
    // Async stage of K^T and V blocks (coalesced on m; LDS side scattered).
    for (int j = 0; j < 16; ++j) {
      const int e = j * 256 + t;
      const int m = e & 63, d = e >> 6;
      const float* g = kb + ((size_t)d * H + h) * N + mb + m;
      async_copy_b32(g, &ldsK[m * LKV + d]);
      const float* gv = vb + ((size_t)d * H + h) * N + mb + m;
      async_copy_b32(gv, &ldsV[d * LKV + m]);
    }
    wait_async_lds();
    __syncthreads();

    for (int sub = 0; sub < 4; ++sub) {
      const int m0 = sub * 16;
      // S^T tile (rows m, cols n) = K^T * Q
      v8f s = {};
#pragma unroll
      for (int kk = 0; kk < 16; ++kk) {
        const float* kp = &ldsK[(m0 + col) * LKV + 4 * kk + 2 * half];
        v2f a; a.x = kp[0]; a.y = kp[1];
        s = WMMA_F32X4(a, Qf[kk], s);
      }
      // Online softmax (per n-column == per lane; halves combined via xor16).
      float tmax = -1e30f;
#pragma unroll
      for (int r = 0; r < 8; ++r) { s[r] *= 0.125f; tmax = fmaxf(tmax, s[r]); }
      tmax = fmaxf(tmax, __shfl_xor(tmax, 16, 32));
      const float m_new = fmaxf(m_run, tmax);
      const float alpha = __expf(m_run - m_new);
      float tsum = 0.0f;
#pragma unroll
      for (int r = 0; r < 8; ++r) {
        const float p = __expf(s[r] - m_new);
        s[r] = p;
        tsum += p;
      }
      tsum += __shfl_xor(tsum, 16, 32);
      l_run = l_run * alpha + tsum;
      m_run = m_new;
#pragma unroll
      for (int dt = 0; dt < 4; ++dt)
#pragma unroll
        for (int r = 0; r < 8; ++r) acc[dt][r] *= alpha;

      // Round-trip P through wave-private LDS to form B fragments.
#pragma unroll
      for (int r = 0; r < 8; ++r)
        Pw[(r + 8 * half) * 17 + col] = s[r];

      // O += V * P   (A = V 16d x 4m, B = P 4m x 16n)
#pragma unroll
      for (int dt = 0; dt < 4; ++dt) {
#pragma unroll
        for (int mk = 0; mk < 4; ++mk) {
          const float* vp = &ldsV[(dt * 16 + col) * LKV + m0 + 4 * mk + 2 * half];
          v2f a; a.x = vp[0]; a.y = vp[1];
          const int mr = 4 * mk + 2 * half;
          v2f bf;
          bf.x = Pw[mr * 17 + col];
          bf.y = Pw[(mr + 1) * 17 + col];
          acc[dt] = WMMA_F32X4(a, bf, acc[dt]);
        }
      }
    }
  }

  const float inv_l = 1.0f / l_run;
  float* ab = attn + (size_t)b * D * N;
#pragma unroll
  for (int dt = 0; dt < 4; ++dt) {
#pragma unroll
    for (int r = 0; r < 8; ++r) {
      const int d = dt * 16 + r + 8 * half;
      ab[((size_t)d * H + h) * N + n0 + col] = acc[dt][r] * inv_l;
    }
  }
}

// ---------------------------------------------------------------------------
// InstanceNorm1d (biased var over N) + ReLU, in place on h (B, 512, N).
// Grid: (512, BATCH), block 256.
// ---------------------------------------------------------------------------
__global__ __launch_bounds__(256)
void instnorm_relu_kernel(float* __restrict__ hbuf)
{
  __shared__ float red[2][8];
  const int t = threadIdx.x;
  const int c = blockIdx.x;
  const int b = blockIdx.y;
  float* p = hbuf + ((size_t)b * D2 + c) * N;

  float s = 0.0f, s2 = 0.0f;
  for (int i = t; i < N; i += 256) { const float x = p[i]; s += x; s2 += x * x; }
#pragma unroll
  for (int off = 16; off > 0; off >>= 1) {
    s  += __shfl_xor(s,  off, 32);
    s2 += __shfl_xor(s2, off, 32);
  }
  const int wave = t >> 5, lane = t & 31;
  if (lane == 0) { red[0][wave] = s; red[1][wave] = s2; }
  __syncthreads();
  float ts = 0.0f, ts2 = 0.0f;
#pragma unroll
  for (int w = 0; w < 8; ++w) { ts += red[0][w]; ts2 += red[1][w]; }
  const float mu   = ts / (float)N;
  const float var  = ts2 / (float)N - mu * mu;
  const float rstd = rsqrtf(var + 1e-5f);
  for (int i = t; i < N; i += 256) {
    const float x = (p[i] - mu) * rstd;
    p[i] = x > 0.0f ? x : 0.0f;
  }
}

// ---------------------------------------------------------------------------
extern "C" void kernel_launch(void* const* d_in, const int* in_sizes, int n_in,
                              void* d_out, int out_size, void* d_ws, size_t ws_size,
                              hipStream_t stream)
{
  const float* x   = (const float*)d_in[0];
  const float* src = (const float*)d_in[1];
  const float* Wq  = (const float*)d_in[2];
  const float* bq  = (const float*)d_in[3];
  const float* Wk  = (const float*)d_in[4];
  const float* bk  = (const float*)d_in[5];
  const float* Wv  = (const float*)d_in[6];
  const float* bv  = (const float*)d_in[7];
  const float* Wm  = (const float*)d_in[8];
  const float* bm  = (const float*)d_in[9];
  const float* W1  = (const float*)d_in[10];
  const float* b1  = (const float*)d_in[11];
  const float* W2  = (const float*)d_in[12];
  const float* b2  = (const float*)d_in[13];
  float* out = (float*)d_out;

  float* ws = (float*)d_ws;
  const size_t sz_dn = (size_t)BATCH * D * N;
  float* qb = ws;
  float* kb = qb + sz_dn;
  float* vb = kb + sz_dn;
  float* at = vb + sz_dn;
  float* ms = at + sz_dn;
  float* hb = ms + sz_dn;                     // BATCH * 512 * N

  const dim3 blk(256);
  // q = Wq x + bq ; k = Wk s + bk ; v = Wv s + bv
  conv1_gemm_kernel<<<dim3(N / 128, D / 16, BATCH), blk, 0, stream>>>(Wq, bq, x,   x,   D,  D,  D,  qb);
  conv1_gemm_kernel<<<dim3(N / 128, D / 16, BATCH), blk, 0, stream>>>(Wk, bk, src, src, D,  D,  D,  kb);
  conv1_gemm_kernel<<<dim3(N / 128, D / 16, BATCH), blk, 0, stream>>>(Wv, bv, src, src, D,  D,  D,  vb);
  // attention
  attention_kernel<<<dim3(N / 128, H, BATCH), blk, 0, stream>>>(qb, kb, vb, at);
  // message = Wm attn + bm
  conv1_gemm_kernel<<<dim3(N / 128, D / 16, BATCH), blk, 0, stream>>>(Wm, bm, at,  at,  D,  D,  D,  ms);
  // h = W1 [x; message] + b1
  conv1_gemm_kernel<<<dim3(N / 128, D2 / 16, BATCH), blk, 0, stream>>>(W1, b1, x,  ms,  D,  D2, D2, hb);
  // instance norm + relu, in place
  instnorm_relu_kernel<<<dim3(D2, BATCH), blk, 0, stream>>>(hb);
  // out = W2 h + b2
  conv1_gemm_kernel<<<dim3(N / 128, D / 16, BATCH), blk, 0, stream>>>(W2, b2, hb,  hb,  D2, D,  D2, out);
}